// RepresentationFunc_31988916420846
// MI455X (gfx1250) — compile-verified
//
#include <hip/hip_runtime.h>

typedef float v2f __attribute__((ext_vector_type(2)));
typedef float v8f __attribute__((ext_vector_type(8)));

// Guaranteed single-instruction non-returning f32 atomic add (tracked by STOREcnt,
// implicitly waited at s_endpgm).
__device__ __forceinline__ void atomic_add_f32_hw(float* p, float v) {
    asm volatile("global_atomic_add_f32 %0, %1, off"
                 :: "v"((unsigned long long)(uintptr_t)p), "v"(v)
                 : "memory");
}

// ---------------- degree / normalization ----------------

__global__ __launch_bounds__(256) void k_deg_init(float* __restrict__ deg, int n) {
    int i = blockIdx.x * 256 + threadIdx.x;
    if (i < n) deg[i] = 1.0f;                       // self loop contributes 1
}

__global__ __launch_bounds__(256) void k_deg_count(const long long* __restrict__ row,
                                                   float* __restrict__ deg, int nE) {
    int e = blockIdx.x * 256 + threadIdx.x;
    if (e < nE) {
        int r = (int)row[e];
        atomic_add_f32_hw(&deg[r], 1.0f);
    }
}

__global__ __launch_bounds__(256) void k_deg_rsqrt(float* __restrict__ deg, int n) {
    int i = blockIdx.x * 256 + threadIdx.x;
    if (i < n) deg[i] = rsqrtf(deg[i]);             // deg -> dis in place
}

// ---------------- concat [x | feat] -> h0 ----------------

__global__ __launch_bounds__(256) void k_concat(const float* __restrict__ x,
                                                const float* __restrict__ feat,
                                                float* __restrict__ h0, int n) {
    int g = blockIdx.x * 256 + threadIdx.x;
    if (g >= n * 16) return;
    int i = g >> 4, f4 = g & 15;
    float4 v;
    if (f4 < 8) v = reinterpret_cast<const float4*>(x)[i * 8 + f4];
    else        v = reinterpret_cast<const float4*>(feat)[i * 8 + (f4 - 8)];
    reinterpret_cast<float4*>(h0)[i * 16 + f4] = v;
}

// ---------------- self-loop term: hnext[i] = dis[i]^2 * lin[i] ----------------

__global__ __launch_bounds__(256) void k_selfloop(const float* __restrict__ lin,
                                                  const float* __restrict__ dis,
                                                  float* __restrict__ hnext, int n) {
    int g = blockIdx.x * 256 + threadIdx.x;
    if (g >= n * 16) return;
    int i = g >> 4, f4 = g & 15;
    float d = dis[i];
    float w = d * d;
    float4 v = reinterpret_cast<const float4*>(lin)[i * 16 + f4];
    v.x *= w; v.y *= w; v.z *= w; v.w *= w;
    reinterpret_cast<float4*>(hnext)[i * 16 + f4] = v;
}

// ---------------- edge scatter: hnext[col] += dis[row]*dis[col] * lin[row] ----------------
// 16 lanes per edge, float4 per lane (coalesced 256B gather per edge, L2-resident atomics)

__global__ __launch_bounds__(256) void k_scatter(const long long* __restrict__ row,
                                                 const long long* __restrict__ col,
                                                 const float* __restrict__ dis,
                                                 const float* __restrict__ lin,
                                                 float* __restrict__ acc, int nE) {
    int g = blockIdx.x * 256 + threadIdx.x;
    int e = g >> 4;
    if (e >= nE) return;
    int f4 = g & 15;
    int r = (int)row[e];
    int c = (int)col[e];
    float w = dis[r] * dis[c];
    float4 v = reinterpret_cast<const float4*>(lin)[r * 16 + f4];
    float* dst = acc + (size_t)c * 64 + (size_t)f4 * 4;
    atomic_add_f32_hw(dst + 0, w * v.x);
    atomic_add_f32_hw(dst + 1, w * v.y);
    atomic_add_f32_hw(dst + 2, w * v.z);
    atomic_add_f32_hw(dst + 3, w * v.w);
}

// ---------------- fp32 WMMA GEMM: out[N,64] = (reluIn? relu(A):A) @ W[64,64] + b ----------------
// Block: 256 threads = 8 waves; each wave owns a 16-row tile (block covers 128 rows).
//
// LDS layouts (wave32, lane = hi*16 + m):
//  - sWt pair-major: sWt[kk*256 + t*64 + lane*2 + j] = W[4kk + 2hi + j][16t + m]
//      -> each B fragment is ONE aligned ds_load_b64; banks = 2*lane (conflict-free).
//  - sA row stride 68 (padded): a fragment at m*68 + 2hi + 4kk, aligned b64;
//      banks = 4m + 2hi (+j) -> conflict-free across the wave.

__global__ __launch_bounds__(256) void k_gemm64(const float* __restrict__ A,
                                                const float* __restrict__ W,
                                                const float* __restrict__ bias,
                                                float* __restrict__ out,
                                                int n, int reluIn, int reluOut) {
    __shared__ float sWt[64 * 64];        // 16 KB, pair-major
    __shared__ float sB[64];
    __shared__ float sA[8][16 * 68];      // 34 KB (8 waves x padded 16x64 tile)

    const int tid  = threadIdx.x;
    const int wave = tid >> 5;
    const int lane = tid & 31;

    // cooperative load of W into pair-major layout (+ bias)
    {
        const float4* W4 = reinterpret_cast<const float4*>(W);
        #pragma unroll
        for (int i = 0; i < 4; ++i) {
            int flat = i * 256 + tid;          // float4 index over W
            int k    = flat >> 4;              // source row (K)
            int c0   = (flat & 15) * 4;        // first of 4 consecutive cols
            float4 v = W4[flat];
            float e[4] = {v.x, v.y, v.z, v.w};
            #pragma unroll
            for (int jj = 0; jj < 4; ++jj) {
                int c = c0 + jj;
                sWt[(k >> 2) * 256 + (c >> 4) * 64 + ((k >> 1) & 1) * 32 + (c & 15) * 2 + (k & 1)]
                    = e[jj];
            }
        }
        if (tid < 16) reinterpret_cast<float4*>(sB)[tid] = reinterpret_cast<const float4*>(bias)[tid];
    }

    // stage this wave's 16x64 A tile (zero-padded past n, optional fused ReLU)
    const int row0 = blockIdx.x * 128 + wave * 16;
    {
        #pragma unroll
        for (int it = 0; it < 8; ++it) {
            int idx = it * 32 + lane;          // 0..255 float4 slots
            int r   = idx >> 4;
            int c4  = idx & 15;
            int gr  = row0 + r;
            float4 v = make_float4(0.f, 0.f, 0.f, 0.f);
            if (gr < n) v = reinterpret_cast<const float4*>(A)[gr * 16 + c4];
            if (reluIn) {
                v.x = fmaxf(v.x, 0.f); v.y = fmaxf(v.y, 0.f);
                v.z = fmaxf(v.z, 0.f); v.w = fmaxf(v.w, 0.f);
            }
            *reinterpret_cast<float4*>(&sA[wave][r * 68 + c4 * 4]) = v;
        }
    }
    __syncthreads();

    v8f acc[4];
    #pragma unroll
    for (int t = 0; t < 4; ++t) acc[t] = (v8f){0.f, 0.f, 0.f, 0.f, 0.f, 0.f, 0.f, 0.f};

    const int m  = lane & 15;
    const int hi = lane >> 4;
    const float* aBase = &sA[wave][m * 68 + hi * 2];   // + kk*4
    const float* bBase = &sWt[lane * 2];               // + kk*256 + t*64

    #pragma unroll
    for (int kk = 0; kk < 16; ++kk) {
        v2f a = *reinterpret_cast<const v2f*>(aBase + kk * 4);
        #pragma unroll
        for (int t = 0; t < 4; ++t) {
            v2f b = *reinterpret_cast<const v2f*>(bBase + kk * 256 + t * 64);
            acc[t] = __builtin_amdgcn_wmma_f32_16x16x4_f32(
                false, a, false, b, (short)0, acc[t], false, false);
        }
    }

    // epilogue: C/D layout -> lane m = column, VGPR v -> row v (+8 for hi half)
    #pragma unroll
    for (int t = 0; t < 4; ++t) {
        const int cb = t * 16 + m;
        const float bb = sB[cb];
        #pragma unroll
        for (int v = 0; v < 8; ++v) {
            int gr = row0 + hi * 8 + v;
            if (gr < n) {
                float val = acc[t][v] + bb;
                if (reluOut) val = fmaxf(val, 0.f);
                out[gr * 64 + cb] = val;
            }
        }
    }
}

// ---------------- launch ----------------

extern "C" void kernel_launch(void* const* d_in, const int* in_sizes, int n_in,
                              void* d_out, int out_size, void* d_ws, size_t ws_size,
                              hipStream_t stream) {
    const float*     x    = (const float*)d_in[0];
    const float*     feat = (const float*)d_in[1];
    const long long* eidx = (const long long*)d_in[2];   // int64 [2, E] flat
    const float*     W1   = (const float*)d_in[3];
    const float*     b1   = (const float*)d_in[4];
    const float*     W2   = (const float*)d_in[5];
    const float*     b2   = (const float*)d_in[6];
    const float*     Wfc  = (const float*)d_in[7];
    const float*     bfc  = (const float*)d_in[8];
    float*           out  = (float*)d_out;

    const int N  = in_sizes[0] / 32;
    const int nE = in_sizes[2] / 2;
    const long long* row = eidx;
    const long long* col = eidx + nE;

    float* ws   = (float*)d_ws;
    size_t disN = ((size_t)N + 63) & ~(size_t)63;        // 256B-aligned slot
    float* dis  = ws;
    float* bufA = ws + disN;                             // N x 64
    float* bufB = bufA + (size_t)N * 64;                 // N x 64
    float* bufC = bufB + (size_t)N * 64;                 // N x 64
    (void)ws_size; (void)n_in; (void)out_size;

    dim3 blk(256);
    int gN   = (N + 255) / 256;
    int gE   = (nE + 255) / 256;
    int gNF4 = (N * 16 + 255) / 256;
    int gE16 = (int)(((long long)nE * 16 + 255) / 256);
    int gG   = (N + 127) / 128;

    // normalization coefficients
    k_deg_init <<<gN,   blk, 0, stream>>>(dis, N);
    k_deg_count<<<gE,   blk, 0, stream>>>(row, dis, nE);
    k_deg_rsqrt<<<gN,   blk, 0, stream>>>(dis, N);

    // h0 = [x | feat]
    k_concat   <<<gNF4, blk, 0, stream>>>(x, feat, bufA, N);

    // layer 1
    k_gemm64   <<<gG,   blk, 0, stream>>>(bufA, W1, b1, bufB, N, 0, 0);
    k_selfloop <<<gNF4, blk, 0, stream>>>(bufB, dis, bufC, N);
    k_scatter  <<<gE16, blk, 0, stream>>>(row, col, dis, bufB, bufC, nE);

    // layer 2 (ReLU of layer-1 output fused into A-load)
    k_gemm64   <<<gG,   blk, 0, stream>>>(bufC, W2, b2, bufA, N, 1, 0);
    k_selfloop <<<gNF4, blk, 0, stream>>>(bufA, dis, bufB, N);
    k_scatter  <<<gE16, blk, 0, stream>>>(row, col, dis, bufA, bufB, nE);

    // final FC (ReLU in, ReLU out)
    k_gemm64   <<<gG,   blk, 0, stream>>>(bufB, Wfc, bfc, out, N, 1, 1);
}